// GraphNeuralNetwork_50491635532438
// MI455X (gfx1250) — compile-verified
//
#include <hip/hip_runtime.h>

typedef float v2f __attribute__((ext_vector_type(2)));
typedef float v8f __attribute__((ext_vector_type(8)));

#define F_DIM   64
#define NCLS    16

// ---------------------------------------------------------------------------
// Transpose X [64, N] (feature-major) -> Xn [N, 64] (node-major, row = 256B)
// ---------------------------------------------------------------------------
__global__ __launch_bounds__(256) void k_transpose(const float* __restrict__ X,
                                                   float* __restrict__ Xn, int n) {
    __shared__ float tile[32][33];
    int nblk = blockIdx.x * 32;     // node offset
    int fblk = blockIdx.y * 32;     // feature offset (0 or 32)
    int tx = threadIdx.x, ty = threadIdx.y;
#pragma unroll
    for (int i = ty; i < 32; i += 8) {
        int nn = nblk + tx;
        tile[i][tx] = (nn < n) ? X[(size_t)(fblk + i) * n + nn] : 0.0f;
    }
    __syncthreads();
#pragma unroll
    for (int i = ty; i < 32; i += 8) {
        int nn = nblk + i;
        if (nn < n) Xn[(size_t)nn * F_DIM + fblk + tx] = tile[tx][i];
    }
}

// ---------------------------------------------------------------------------
// Edge scatter: agg[dst] += feat[src] * val   (16 lanes per edge, float4 each)
// Hardware fp32 global atomics (no-return) -> fire-and-forget into L2.
// ---------------------------------------------------------------------------
__global__ __launch_bounds__(256) void k_scatter(const float* __restrict__ feat,
                                                 const int*   __restrict__ src,
                                                 const int*   __restrict__ dst,
                                                 const float* __restrict__ val,
                                                 float* __restrict__ agg, int n_edges) {
    int t = blockIdx.x * blockDim.x + threadIdx.x;
    int e = t >> 4;
    if (e >= n_edges) return;
    int fo = (t & 15) << 2;                 // feature offset (0..60, step 4)
    int s = src[e], d = dst[e];
    float v = val[e];
    const float4 x = *reinterpret_cast<const float4*>(feat + (size_t)s * F_DIM + fo);
    float* a = agg + (size_t)d * F_DIM + fo;
    unsafeAtomicAdd(a + 0, x.x * v);
    unsafeAtomicAdd(a + 1, x.y * v);
    unsafeAtomicAdd(a + 2, x.z * v);
    unsafeAtomicAdd(a + 3, x.w * v);
}

// ---------------------------------------------------------------------------
// Layer 1: h = relu(Xn @ Wu + agg @ Wv + b)   [N,64]x[64,64] via WMMA f32
// Block = 256 thr (8 waves): 2 node-tiles x 4 feature-tiles of 16x16 each.
// ---------------------------------------------------------------------------
__global__ __launch_bounds__(256) void k_layer1(const float* __restrict__ Xn,
                                                const float* __restrict__ agg,
                                                const float* __restrict__ Wu,
                                                const float* __restrict__ Wv,
                                                const float* __restrict__ bias,
                                                float* __restrict__ h, int n) {
    int wave = threadIdx.x >> 5;
    int lane = threadIdx.x & 31;
    int m0 = blockIdx.x * 32 + (wave >> 2) * 16;   // node tile
    int n0 = (wave & 3) * 16;                      // out-feature tile
    int row  = lane & 15;
    int hi   = lane >> 4;
    int kofs = hi * 2;
    int mr = m0 + row; if (mr >= n) mr = n - 1;    // clamp (loads only)

    v8f c = {};
#pragma unroll
    for (int k = 0; k < F_DIM; k += 4) {
        v2f a1 = *reinterpret_cast<const v2f*>(Xn + (size_t)mr * F_DIM + k + kofs);
        v2f b1 = { Wu[(k + kofs) * F_DIM + n0 + row],
                   Wu[(k + kofs + 1) * F_DIM + n0 + row] };
        c = __builtin_amdgcn_wmma_f32_16x16x4_f32(false, a1, false, b1,
                                                  (short)0, c, false, false);
        v2f a2 = *reinterpret_cast<const v2f*>(agg + (size_t)mr * F_DIM + k + kofs);
        v2f b2 = { Wv[(k + kofs) * F_DIM + n0 + row],
                   Wv[(k + kofs + 1) * F_DIM + n0 + row] };
        c = __builtin_amdgcn_wmma_f32_16x16x4_f32(false, a2, false, b2,
                                                  (short)0, c, false, false);
    }
    float bv = bias[n0 + row];
#pragma unroll
    for (int g = 0; g < 8; ++g) {
        int m = m0 + g + 8 * hi;
        if (m < n) {
            float x = c[g] + bv;
            h[(size_t)m * F_DIM + n0 + row] = x > 0.0f ? x : 0.0f;
        }
    }
}

// ---------------------------------------------------------------------------
// Layer 2: out = log_softmax(h @ Wu2 + agg @ Wv2 + b2)  [N,64]x[64,16]
// One 16x16 tile per wave; row-wise softmax via 16-wide shfl_xor butterflies.
// ---------------------------------------------------------------------------
__global__ __launch_bounds__(256) void k_layer2(const float* __restrict__ H,
                                                const float* __restrict__ agg,
                                                const float* __restrict__ Wu,
                                                const float* __restrict__ Wv,
                                                const float* __restrict__ bias,
                                                float* __restrict__ out, int n) {
    int wave = threadIdx.x >> 5;
    int lane = threadIdx.x & 31;
    int tile = blockIdx.x * 8 + wave;
    int ntiles = (n + 15) / 16;
    if (tile >= ntiles) return;                    // wave-uniform: EXEC stays full
    int m0 = tile * 16;
    int row  = lane & 15;
    int hi   = lane >> 4;
    int kofs = hi * 2;
    int mr = m0 + row; if (mr >= n) mr = n - 1;

    v8f c = {};
#pragma unroll
    for (int k = 0; k < F_DIM; k += 4) {
        v2f a1 = *reinterpret_cast<const v2f*>(H + (size_t)mr * F_DIM + k + kofs);
        v2f b1 = { Wu[(k + kofs) * NCLS + row],
                   Wu[(k + kofs + 1) * NCLS + row] };
        c = __builtin_amdgcn_wmma_f32_16x16x4_f32(false, a1, false, b1,
                                                  (short)0, c, false, false);
        v2f a2 = *reinterpret_cast<const v2f*>(agg + (size_t)mr * F_DIM + k + kofs);
        v2f b2 = { Wv[(k + kofs) * NCLS + row],
                   Wv[(k + kofs + 1) * NCLS + row] };
        c = __builtin_amdgcn_wmma_f32_16x16x4_f32(false, a2, false, b2,
                                                  (short)0, c, false, false);
    }
    float bv = bias[row];
#pragma unroll
    for (int g = 0; g < 8; ++g) {
        float x = c[g] + bv;                       // row m = m0 + g + 8*hi, col = row
        float mx = x;
#pragma unroll
        for (int off = 8; off >= 1; off >>= 1)
            mx = fmaxf(mx, __shfl_xor(mx, off, 16));
        float s = expf(x - mx);
#pragma unroll
        for (int off = 8; off >= 1; off >>= 1)
            s += __shfl_xor(s, off, 16);
        int m = m0 + g + 8 * hi;
        if (m < n) out[(size_t)m * NCLS + row] = x - mx - logf(s);
    }
}

// ---------------------------------------------------------------------------
extern "C" void kernel_launch(void* const* d_in, const int* in_sizes, int n_in,
                              void* d_out, int out_size, void* d_ws, size_t ws_size,
                              hipStream_t stream) {
    const float* X   = (const float*)d_in[0];
    const int*   src = (const int*)  d_in[1];
    const int*   dst = (const int*)  d_in[2];
    const float* val = (const float*)d_in[3];
    const float* Wu1 = (const float*)d_in[4];
    const float* Wv1 = (const float*)d_in[5];
    const float* b1  = (const float*)d_in[6];
    const float* Wu2 = (const float*)d_in[7];
    const float* Wv2 = (const float*)d_in[8];
    const float* b2  = (const float*)d_in[9];

    const int n  = in_sizes[0] / F_DIM;   // X is [64, N]
    const int ne = in_sizes[1];

    float* Xn  = (float*)d_ws;                     // [N,64]
    float* agg = Xn  + (size_t)n * F_DIM;          // [N,64] (reused per layer)
    float* h   = agg + (size_t)n * F_DIM;          // [N,64]

    // 1. X^T -> Xn (coalesced node-major rows)
    dim3 tb(32, 8), tg((n + 31) / 32, 2);
    k_transpose<<<tg, tb, 0, stream>>>(X, Xn, n);

    // 2. layer-1 aggregation
    hipMemsetAsync(agg, 0, (size_t)n * F_DIM * sizeof(float), stream);
    int sblocks = (int)(((size_t)ne * 16 + 255) / 256);
    k_scatter<<<sblocks, 256, 0, stream>>>(Xn, src, dst, val, agg, ne);

    // 3. layer-1 dense: h = relu(Xn@Wu1 + agg@Wv1 + b1)
    k_layer1<<<(n + 31) / 32, 256, 0, stream>>>(Xn, agg, Wu1, Wv1, b1, h, n);

    // 4. layer-2 aggregation
    hipMemsetAsync(agg, 0, (size_t)n * F_DIM * sizeof(float), stream);
    k_scatter<<<sblocks, 256, 0, stream>>>(h, src, dst, val, agg, ne);

    // 5. layer-2 dense + fused log_softmax -> d_out [N,16]
    int ntiles = (n + 15) / 16;
    k_layer2<<<(ntiles + 7) / 8, 256, 0, stream>>>(h, agg, Wu2, Wv2, b2,
                                                   (float*)d_out, n);
}